// LSTM_rater_83399674953988
// MI455X (gfx1250) — compile-verified
//
#include <hip/hip_runtime.h>
#include <hip/hip_bf16.h>
#include <math.h>

#define IN_CH   2048
#define HIDDEN  100
#define GATES   (4 * HIDDEN)   // 400
#define NSLICE  2048
#define ITER_LIMIT 128
#define WHH_ELEMS (GATES * HIDDEN)          // 40000 floats
#define WHH_BYTES (WHH_ELEMS * sizeof(float))  // 160000 B < 320 KB WGP LDS

typedef float v2f __attribute__((ext_vector_type(2)));
typedef float v8f __attribute__((ext_vector_type(8)));

// ---------------------------------------------------------------------------
// Kernel 1: x_gates[s][j] = sum_ch x[ch][s] * W_ih[j][ch] + b_ih[j]
//   M = slices (2048), N = gates (400), K = IN_CH (2048)
//   fp32 WMMA 16x16x4 (full reference precision; the GEMM is cache-BW bound,
//   so bf16 would gain nothing). One wave per 16x16 tile, 512 WMMAs per tile.
// ---------------------------------------------------------------------------
__global__ __launch_bounds__(256) void lstm_xgates_gemm_f32wmma(
    const float* __restrict__ x,
    const float* __restrict__ W_ih,
    const float* __restrict__ b_ih,
    float* __restrict__ xg) {
  const int wave  = (blockIdx.x * 256 + (int)threadIdx.x) >> 5;  // 0..3199
  const int lane  = threadIdx.x & 31;
  const int half  = lane >> 4;     // 0: K base 0, 1: K base 2
  const int l16   = lane & 15;
  const int tileM = wave & 127;    // 128 slice tiles
  const int tileN = wave >> 7;     // 25 gate tiles

  const int m0 = tileM * 16 + l16;           // slice index for A loads (lane = M)
  const int n0 = tileN * 16 + l16;           // gate  index for B loads (lane = N)
  const float* __restrict__ wrow = W_ih + (size_t)n0 * IN_CH;

  v8f acc = {};
#pragma unroll 8
  for (int k0 = 0; k0 < IN_CH; k0 += 4) {
    const int ka = k0 + 2 * half;            // this half-wave's K pair
    v2f a, b;
    // A: 16x4 fp32, A[m][k] = x[k*NSLICE + m] ; lanes 0-15 coalesced over m
    a.x = x[(size_t)(ka + 0) * NSLICE + m0];
    a.y = x[(size_t)(ka + 1) * NSLICE + m0];
    // B: 4x16 fp32, B[k][n] = W_ih[n][k] ; two consecutive K per lane -> b64
    const float2 wv = *(const float2*)(wrow + ka);
    b.x = wv.x;
    b.y = wv.y;
    acc = __builtin_amdgcn_wmma_f32_16x16x4_f32(
        /*neg_a=*/false, a, /*neg_b=*/false, b,
        /*c_mod=*/(short)0, acc, /*reuse_a=*/false, /*reuse_b=*/false);
  }

  // D layout: VGPR r -> row (r + 8*half), lane l16 -> col
  const float bias = b_ih[n0];
#pragma unroll
  for (int r = 0; r < 8; ++r) {
    const int s = tileM * 16 + r + 8 * half;
    xg[(size_t)s * GATES + n0] = acc[r] + bias;
  }
}

// ---------------------------------------------------------------------------
// Kernel 2: sequential LSTM (phase 1 full scan + phase 2 data-dependent walk)
//   Single persistent workgroup of 512 threads (16 wave32s, one WGP).
//   W_hh (160 KB) is staged ONCE into CDNA5's 320 KB LDS so every serial
//   step's matvec runs at LDS latency instead of L2 latency.
//   Row stride 100 dwords: gcd(100,64)=4 -> b128 reads hit all 64 banks
//   exactly twice (minimum possible), so no padding needed.
// ---------------------------------------------------------------------------
__device__ __forceinline__ float sigmoid_f(float v) {
  return 1.0f / (1.0f + __expf(-v));
}

__global__ __launch_bounds__(512) void lstm_recurrent_seq(
    const float* __restrict__ xg,
    const float* __restrict__ W_hh,
    const float* __restrict__ b_hh,
    const float* __restrict__ W_fc,
    const float* __restrict__ b_fc,
    float* __restrict__ out) {
  extern __shared__ float sW[];            // [GATES * HIDDEN] dynamic LDS
  __shared__ float sh_h[HIDDEN];
  __shared__ float sh_c[HIDDEN];
  __shared__ float sh_z[GATES];
  __shared__ float sh_o[3];
  __shared__ float sh_olast[3];
  __shared__ int sh_idx, sh_consec, sh_done;

  const int j = threadIdx.x;

  // One-time stage of W_hh into LDS (coalesced b32 stream; cost amortized
  // over 2176 serial steps).
  for (int i = j; i < WHH_ELEMS; i += 512) sW[i] = W_hh[i];

  if (j < HIDDEN) { sh_h[j] = 0.0f; sh_c[j] = 0.0f; }
  if (j == 0) { sh_idx = NSLICE / 2; sh_consec = 0; sh_done = 0; }
  __syncthreads();

  // One LSTM cell step on slice s (PyTorch gate order i,f,g,o).
  auto cell = [&](int s, int s_next) {
    if (j < GATES) {
      // Pull the upcoming slice's x_gates row toward the WGP ahead of time
      // (gfx1250 global_prefetch_b8; speculative, OOB-safe but clamped).
      __builtin_prefetch(xg + (size_t)s_next * GATES + j, 0, 1);
      float acc = xg[(size_t)s * GATES + j] + b_hh[j];
      const float4* __restrict__ wrow = (const float4*)(sW + j * HIDDEN);
#pragma unroll
      for (int k4 = 0; k4 < HIDDEN / 4; ++k4) {
        const float4 w  = wrow[k4];
        const float4 hv = *(const float4*)(sh_h + 4 * k4);
        acc = fmaf(w.x, hv.x, acc);
        acc = fmaf(w.y, hv.y, acc);
        acc = fmaf(w.z, hv.z, acc);
        acc = fmaf(w.w, hv.w, acc);
      }
      sh_z[j] = acc;
    }
    __syncthreads();
    if (j < HIDDEN) {
      const float zi = sh_z[j];
      const float zf = sh_z[HIDDEN + j];
      const float zg = sh_z[2 * HIDDEN + j];
      const float zo = sh_z[3 * HIDDEN + j];
      const float cn = sigmoid_f(zf) * sh_c[j] + sigmoid_f(zi) * tanhf(zg);
      const float hn = sigmoid_f(zo) * tanhf(cn);
      sh_c[j] = cn;
      sh_h[j] = hn;
    }
    __syncthreads();
  };

  // Phase 1: full pass over all slices; only final (h, c) kept.
  for (int s = 0; s < NSLICE; ++s) {
    const int s_next = (s + 8 < NSLICE) ? s + 8 : NSLICE - 1;
    cell(s, s_next);
  }

  // Phase 2: masked/while scan starting at the middle slice.
  for (int t = 0; t < ITER_LIMIT; ++t) {
    if (sh_done) break;           // uniform: shared, written before a barrier
    const int idx = sh_idx;
    cell(idx, idx);
    if (j < 3) {
      float acc = b_fc[j];
#pragma unroll 10
      for (int k = 0; k < HIDDEN; ++k)
        acc = fmaf(W_fc[j * HIDDEN + k], sh_h[k], acc);
      sh_o[j] = acc;
    }
    __syncthreads();
    if (j == 0) {
      const float o1 = sh_o[1], o2 = sh_o[2];
      sh_olast[0] = sh_o[0]; sh_olast[1] = o1; sh_olast[2] = o2;
      int idx2 = (o2 > 0.0f) ? idx + 1 : idx - 1;
      idx2 = ((idx2 % NSLICE) + NSLICE) % NSLICE;   // jnp.mod semantics
      const int consec2 = (o1 > 0.0f) ? sh_consec + 1 : 0;
      sh_consec = consec2;
      sh_idx = idx2;
      if (consec2 > 3) sh_done = 1;
    }
    __syncthreads();
  }

  if (j < 2) out[j] = sh_olast[j];
}

// ---------------------------------------------------------------------------
extern "C" void kernel_launch(void* const* d_in, const int* in_sizes, int n_in,
                              void* d_out, int out_size, void* d_ws, size_t ws_size,
                              hipStream_t stream) {
  (void)in_sizes; (void)n_in; (void)out_size; (void)ws_size;
  const float* x    = (const float*)d_in[0];  // [1, IN_CH, NSLICE]
  const float* W_ih = (const float*)d_in[1];  // [4H, IN_CH]
  const float* W_hh = (const float*)d_in[2];  // [4H, HIDDEN]
  const float* b_ih = (const float*)d_in[3];  // [4H]
  const float* b_hh = (const float*)d_in[4];  // [4H]
  const float* W_fc = (const float*)d_in[5];  // [3, HIDDEN]
  const float* b_fc = (const float*)d_in[6];  // [3]
  float* out = (float*)d_out;                 // [1, 2]
  float* xg  = (float*)d_ws;                  // [NSLICE, 4H] = 3.28 MB scratch

  // 3200 tiles (128 slice-tiles x 25 gate-tiles), 1 wave per tile, 8 waves/block.
  lstm_xgates_gemm_f32wmma<<<400, 256, 0, stream>>>(x, W_ih, b_ih, xg);
  // Single persistent workgroup; 160 KB dynamic LDS for W_hh (CDNA5: 320 KB/WGP).
  lstm_recurrent_seq<<<1, 512, WHH_BYTES, stream>>>(xg, W_hh, b_hh, W_fc, b_fc, out);
}